// MSDformer_13529146982472
// MI455X (gfx1250) — compile-verified
//
#include <hip/hip_runtime.h>
#include <hip/hip_bf16.h>
#include <math.h>

// Problem constants (match reference)
#define BB    4
#define CC    256
#define HH    64
#define WWID  64
#define HWXY  4096      // H*W
#define NTOK  16384     // B*H*W tokens
#define P2    64        // windows per image
#define W2    64        // tokens per window
#define HEADS 8
#define HD    32
#define SCL   0.0625f   // C^-0.5

typedef __attribute__((ext_vector_type(2))) float v2f;
typedef __attribute__((ext_vector_type(8))) float v8f;

// D = A(16x4) * B(4x16) + C, fp32 WMMA (CDNA5 V_WMMA_F32_16X16X4_F32)
__device__ __forceinline__ v8f wmma4(v2f a, v2f b, v8f c) {
  return __builtin_amdgcn_wmma_f32_16x16x4_f32(false, a, false, b, (short)0, c,
                                               false, false);
}

// ---------------------------------------------------------------------------
// GEMM over image layout: Out[b][m][n] = act(sum_c Wt[m][c]*X[b][c][n] + bias)
// X, Out: (B, 256, 4096). MODE: 0 none, 1 tanh, 2 sigmoid.
// One wave -> one 16(M) x 64(N) strip (4 WMMA tiles), A fragment reused 4x.
// Waves: B * 16(mt) * 64(ng) = 4096 -> 512 blocks of 256 threads.
// ---------------------------------------------------------------------------
template <int MODE>
__global__ __launch_bounds__(256) void gemm_chw(const float* __restrict__ Wt,
                                                const float* __restrict__ bias,
                                                const float* __restrict__ X,
                                                float* __restrict__ Out) {
  const int wave = (blockIdx.x * blockDim.x + threadIdx.x) >> 5;
  const int lane = threadIdx.x & 31;
  const int ng = wave & 63;           // 64-column group within batch
  const int mt = (wave >> 6) & 15;    // output-channel tile
  const int b  = wave >> 10;
  const int half = lane >> 4, rc = lane & 15;

  const float* Arow = Wt + (size_t)(mt * 16 + rc) * CC;
  const float* Bcol = X + (size_t)b * CC * HWXY + ng * 64 + rc;

  v8f acc0 = {0.f,0.f,0.f,0.f,0.f,0.f,0.f,0.f};
  v8f acc1 = acc0, acc2 = acc0, acc3 = acc0;
  for (int k0 = 0; k0 < CC; k0 += 4) {
    v2f a;
    a.x = Arow[k0 + 2 * half];
    a.y = Arow[k0 + 2 * half + 1];
    const float* B0 = Bcol + (size_t)(k0 + 2 * half) * HWXY;
    const float* B1 = Bcol + (size_t)(k0 + 2 * half + 1) * HWXY;
    v2f b0, b1, b2, b3;
    b0.x = B0[0];  b0.y = B1[0];
    b1.x = B0[16]; b1.y = B1[16];
    b2.x = B0[32]; b2.y = B1[32];
    b3.x = B0[48]; b3.y = B1[48];
    acc0 = wmma4(a, b0, acc0);
    acc1 = wmma4(a, b1, acc1);
    acc2 = wmma4(a, b2, acc2);
    acc3 = wmma4(a, b3, acc3);
  }
  float* Ob = Out + (size_t)b * CC * HWXY + ng * 64 + rc;
#pragma unroll
  for (int v = 0; v < 8; v++) {
    int m = mt * 16 + v + 8 * half;
    float bs = bias[m];
    float v0 = acc0[v] + bs, v1 = acc1[v] + bs, v2 = acc2[v] + bs,
          v3 = acc3[v] + bs;
    if (MODE == 1) {
      v0 = tanhf(v0); v1 = tanhf(v1); v2 = tanhf(v2); v3 = tanhf(v3);
    }
    if (MODE == 2) {
      v0 = 1.f / (1.f + expf(-v0)); v1 = 1.f / (1.f + expf(-v1));
      v2 = 1.f / (1.f + expf(-v2)); v3 = 1.f / (1.f + expf(-v3));
    }
    float* row = Ob + (size_t)m * HWXY;
    row[0] = v0; row[16] = v1; row[32] = v2; row[48] = v3;
  }
}

// ---------------------------------------------------------------------------
// headflip(scan(headflip)) == direction-split scan: c<128 backward, c>=128 fwd
// ---------------------------------------------------------------------------
__global__ void scan_kernel(const float* __restrict__ Z,
                            const float* __restrict__ F,
                            float* __restrict__ Y) {
  int i = blockIdx.x * blockDim.x + threadIdx.x;  // B*C*W = 65536
  if (i >= BB * CC * WWID) return;
  int w = i & 63, c = (i >> 6) & 255, b = i >> 14;
  size_t base = ((size_t)(b * CC + c) * HH) * WWID + w;
  float h = 0.f;
  if (c < 128) {
    for (int y = HH - 1; y >= 0; y--) {
      size_t o = base + (size_t)y * WWID;
      float z = Z[o], f = F[o];
      h = f * h + (1.f - f) * z;
      Y[o] = h;
    }
  } else {
    for (int y = 0; y < HH; y++) {
      size_t o = base + (size_t)y * WWID;
      float z = Z[o], f = F[o];
      h = f * h + (1.f - f) * z;
      Y[o] = h;
    }
  }
}

// token t -> image coords
__device__ __forceinline__ void tok2img(int t, int& b, int& h, int& w) {
  b = t >> 12;
  int p = (t >> 6) & 63, q = t & 63;
  h = ((p >> 3) << 3) + (q >> 3);
  w = ((p & 7) << 3) + (q & 7);
}

// ---------------------------------------------------------------------------
// QKV GEMM: QKV[t][m] = sum_c qkv_w[m][c] * Yimg[b][c][h][w] + qkv_b[m]
// M=768, K=256, N=16384 tokens. Token-major output (NTOK x 768).
// One wave -> 16(M) x 64(token) strip. Waves: 48 * 256 = 12288 -> 1536 blocks.
// ---------------------------------------------------------------------------
__global__ __launch_bounds__(256) void gemm_qkv(const float* __restrict__ Wt,
                                                const float* __restrict__ bias,
                                                const float* __restrict__ Y,
                                                float* __restrict__ QKV) {
  const int wave = (blockIdx.x * blockDim.x + threadIdx.x) >> 5;
  const int lane = threadIdx.x & 31;
  const int ng = wave & 255;   // token group of 64
  const int mt = wave >> 8;    // 0..47
  const int half = lane >> 4, rc = lane & 15;

  const float* Arow = Wt + (size_t)(mt * 16 + rc) * CC;
  int tj[4];
  const float* Bcol[4];
#pragma unroll
  for (int j = 0; j < 4; j++) {
    tj[j] = ng * 64 + j * 16 + rc;
    int b, h, w;
    tok2img(tj[j], b, h, w);
    Bcol[j] = Y + (size_t)b * CC * HWXY + h * WWID + w;
  }

  v8f acc[4];
#pragma unroll
  for (int j = 0; j < 4; j++) acc[j] = (v8f){0.f,0.f,0.f,0.f,0.f,0.f,0.f,0.f};

  for (int k0 = 0; k0 < CC; k0 += 4) {
    v2f a;
    a.x = Arow[k0 + 2 * half];
    a.y = Arow[k0 + 2 * half + 1];
    size_t o0 = (size_t)(k0 + 2 * half) * HWXY;
    size_t o1 = (size_t)(k0 + 2 * half + 1) * HWXY;
#pragma unroll
    for (int j = 0; j < 4; j++) {
      v2f bb;
      bb.x = Bcol[j][o0];
      bb.y = Bcol[j][o1];
      acc[j] = wmma4(a, bb, acc[j]);
    }
  }
#pragma unroll
  for (int j = 0; j < 4; j++) {
    float* Oc = QKV + (size_t)tj[j] * 768;
#pragma unroll
    for (int v = 0; v < 8; v++) {
      int m = mt * 16 + v + 8 * half;
      Oc[m] = acc[j][v] + bias[m];
    }
  }
}

// window means of q and k
__global__ void winmean(const float* __restrict__ QKV, float* __restrict__ qwin,
                        float* __restrict__ kwin) {
  int i = blockIdx.x * blockDim.x + threadIdx.x;  // B*P2*C = 65536
  if (i >= BB * P2 * CC) return;
  int c = i & 255, p = (i >> 8) & 63, b = i >> 14;
  const float* base = QKV + ((size_t)(b * P2 + p) * W2) * 768;
  float sq = 0.f, sk = 0.f;
  for (int j = 0; j < W2; j++) {
    sq += base[j * 768 + c];
    sk += base[j * 768 + 256 + c];
  }
  qwin[i] = sq * (1.f / 64.f);
  kwin[i] = sk * (1.f / 64.f);
}

// logits + top-4 window selection (matches lax.top_k tie-breaking)
__global__ void topk_kernel(const float* __restrict__ qwin,
                            const float* __restrict__ kwin,
                            int* __restrict__ idx) {
  int bp = blockIdx.x;        // b*64+p
  int b = bp >> 6;
  int j = threadIdx.x;        // candidate window 0..63
  __shared__ float lg[64];
  const float* qp = qwin + (size_t)bp * CC;
  const float* kj = kwin + (size_t)(b * P2 + j) * CC;
  float s = 0.f;
  for (int c = 0; c < CC; c++) s += qp[c] * kj[c];
  lg[j] = s * SCL;
  __syncthreads();
  if (j == 0) {
    for (int t = 0; t < 4; t++) {
      float best = -3.4e38f; int bi = 0;
      for (int m = 0; m < 64; m++)
        if (lg[m] > best) { best = lg[m]; bi = m; }
      idx[bp * 4 + t] = bi;
      lg[bi] = -3.4e38f;
    }
  }
}

// ---------------------------------------------------------------------------
// Attention: one block (8 waves) per (b, window, head).
// Phase 1: each wave owns one M-tile and 8 N-tiles -> q fragment reused 8x.
// ---------------------------------------------------------------------------
__global__ __launch_bounds__(256) void attn_kernel(const float* __restrict__ QKV,
                                                   const int* __restrict__ idxb,
                                                   float* __restrict__ OutTok) {
  const int bp = blockIdx.x >> 3;     // b*64+p
  const int headi = blockIdx.x & 7;
  const int b = bp >> 6;
  __shared__ float S[64 * 256];       // 64KB score tile

  const int tid = threadIdx.x;
  const int wave = tid >> 5, lane = tid & 31;
  const int half = lane >> 4, rc = lane & 15;
  const int qbase = bp * 64;          // first token of this window
  const int co = headi * HD;          // head channel offset

  const int sel = (idxb[bp * 4]) | (idxb[bp * 4 + 1] << 8) |
                  (idxb[bp * 4 + 2] << 16) | (idxb[bp * 4 + 3] << 24);
  // key (0..255) -> global token of gathered k/v
#define KEYTOK(key) ((b << 12) + ((((sel) >> (((key) >> 6) << 3)) & 63) << 6) + ((key) & 63))

  // Phase 1: S = (q*SCL) k^T ; M=64, N=256, K=32.
  {
    const int mt = wave >> 1;          // 0..3
    const int n0 = (wave & 1) * 8;     // first of 8 n-tiles
    const float* Arow = QKV + (size_t)(qbase + mt * 16 + rc) * 768 + co;
    const float* Bcol[8];
#pragma unroll
    for (int j = 0; j < 8; j++) {
      int key = (n0 + j) * 16 + rc;
      Bcol[j] = QKV + (size_t)KEYTOK(key) * 768 + 256 + co;
    }
    v8f acc[8];
#pragma unroll
    for (int j = 0; j < 8; j++) acc[j] = (v8f){0.f,0.f,0.f,0.f,0.f,0.f,0.f,0.f};
#pragma unroll
    for (int k0 = 0; k0 < HD; k0 += 4) {
      v2f a;
      a.x = Arow[k0 + 2 * half] * SCL;
      a.y = Arow[k0 + 2 * half + 1] * SCL;
#pragma unroll
      for (int j = 0; j < 8; j++) {
        v2f bb;
        bb.x = Bcol[j][k0 + 2 * half];
        bb.y = Bcol[j][k0 + 2 * half + 1];
        acc[j] = wmma4(a, bb, acc[j]);
      }
    }
#pragma unroll
    for (int j = 0; j < 8; j++)
#pragma unroll
      for (int v = 0; v < 8; v++)
        S[(mt * 16 + v + 8 * half) * 256 + (n0 + j) * 16 + rc] = acc[j][v];
  }
  __syncthreads();

  // Phase 2: row softmax (64 rows, one thread per row)
  if (tid < 64) {
    float* row = S + tid * 256;
    float mx = row[0];
    for (int k = 1; k < 256; k++) mx = fmaxf(mx, row[k]);
    float sum = 0.f;
    for (int k = 0; k < 256; k++) { float e = expf(row[k] - mx); row[k] = e; sum += e; }
    float inv = 1.f / sum;
    for (int k = 0; k < 256; k++) row[k] *= inv;
  }
  __syncthreads();

  // Phase 3: O = S(64x256) @ v(256x32) -> 8 tiles, one per wave
  {
    const int mt = wave >> 1, ntl = wave & 1;
    const float* Arow = S + (size_t)(mt * 16 + rc) * 256;
    v8f acc = {0.f,0.f,0.f,0.f,0.f,0.f,0.f,0.f};
    for (int k0 = 0; k0 < 256; k0 += 4) {
      v2f a, bb;
      a.x = Arow[k0 + 2 * half];
      a.y = Arow[k0 + 2 * half + 1];
      int key0 = k0 + 2 * half;
      bb.x = QKV[(size_t)KEYTOK(key0) * 768 + 512 + co + ntl * 16 + rc];
      bb.y = QKV[(size_t)KEYTOK(key0 + 1) * 768 + 512 + co + ntl * 16 + rc];
      acc = wmma4(a, bb, acc);
    }
#pragma unroll
    for (int v = 0; v < 8; v++) {
      int row = mt * 16 + v + 8 * half;
      OutTok[(size_t)(qbase + row) * CC + co + ntl * 16 + rc] = acc[v];
    }
  }
#undef KEYTOK
}

// depthwise 3x3 LePE on v (read from QKV) added in place to OutTok
__global__ void lepe_add(const float* __restrict__ QKV,
                         const float* __restrict__ lw,
                         const float* __restrict__ lb,
                         float* __restrict__ OutTok) {
  int i = blockIdx.x * blockDim.x + threadIdx.x;  // NTOK*C
  if (i >= NTOK * CC) return;
  int c = i & 255, t = i >> 8;
  int b, h, w;
  tok2img(t, b, h, w);
  float s = lb[c];
  for (int dy = -1; dy <= 1; dy++) {
    int hy = h + dy;
    if (hy < 0 || hy >= HH) continue;
    for (int dx = -1; dx <= 1; dx++) {
      int wx = w + dx;
      if (wx < 0 || wx >= WWID) continue;
      int pp = ((hy >> 3) << 3) + (wx >> 3);
      int qq = ((hy & 7) << 3) + (wx & 7);
      int tt = (b << 12) + (pp << 6) + qq;
      s += lw[c * 9 + (dy + 1) * 3 + (dx + 1)] * QKV[(size_t)tt * 768 + 512 + c];
    }
  }
  OutTok[i] += s;
}

// ---------------------------------------------------------------------------
// Output projection: d_out[b][m][h][w] = sum_c wo[m][c]*U[t][c] + wo_b[m]
// One wave -> 16(M) x 64(token). Waves: 16 * 256 = 4096 -> 512 blocks.
// ---------------------------------------------------------------------------
__global__ __launch_bounds__(256) void gemm_out(const float* __restrict__ Wt,
                                                const float* __restrict__ bias,
                                                const float* __restrict__ U,
                                                float* __restrict__ Out) {
  const int wave = (blockIdx.x * blockDim.x + threadIdx.x) >> 5;
  const int lane = threadIdx.x & 31;
  const int ng = wave & 255;
  const int mt = wave >> 8;  // 0..15
  const int half = lane >> 4, rc = lane & 15;

  const float* Arow = Wt + (size_t)(mt * 16 + rc) * CC;
  const float* Bcol[4];
  size_t obase[4];
#pragma unroll
  for (int j = 0; j < 4; j++) {
    int t = ng * 64 + j * 16 + rc;
    int b, h, w;
    tok2img(t, b, h, w);
    Bcol[j] = U + (size_t)t * CC;
    obase[j] = (size_t)b * CC * HWXY + (size_t)h * WWID + w;
  }

  v8f acc[4];
#pragma unroll
  for (int j = 0; j < 4; j++) acc[j] = (v8f){0.f,0.f,0.f,0.f,0.f,0.f,0.f,0.f};

  for (int k0 = 0; k0 < CC; k0 += 4) {
    v2f a;
    a.x = Arow[k0 + 2 * half];
    a.y = Arow[k0 + 2 * half + 1];
#pragma unroll
    for (int j = 0; j < 4; j++) {
      v2f bb;
      bb.x = Bcol[j][k0 + 2 * half];
      bb.y = Bcol[j][k0 + 2 * half + 1];
      acc[j] = wmma4(a, bb, acc[j]);
    }
  }
#pragma unroll
  for (int j = 0; j < 4; j++)
#pragma unroll
    for (int v = 0; v < 8; v++) {
      int m = mt * 16 + v + 8 * half;
      Out[obase[j] + (size_t)m * HWXY] = acc[j][v] + bias[m];
    }
}

// ---------------------------------------------------------------------------
extern "C" void kernel_launch(void* const* d_in, const int* in_sizes, int n_in,
                              void* d_out, int out_size, void* d_ws,
                              size_t ws_size, hipStream_t stream) {
  const float* x      = (const float*)d_in[0];
  const float* w1a    = (const float*)d_in[1];
  const float* b1a    = (const float*)d_in[2];
  const float* w2a    = (const float*)d_in[3];
  const float* b2a    = (const float*)d_in[4];
  const float* w1b    = (const float*)d_in[5];
  const float* b1b    = (const float*)d_in[6];
  const float* w2b    = (const float*)d_in[7];
  const float* b2b    = (const float*)d_in[8];
  const float* qkv_w  = (const float*)d_in[9];
  const float* qkv_b  = (const float*)d_in[10];
  const float* wo_w   = (const float*)d_in[11];
  const float* wo_b   = (const float*)d_in[12];
  const float* lepe_w = (const float*)d_in[13];
  const float* lepe_b = (const float*)d_in[14];
  float* out = (float*)d_out;

  // workspace layout (floats), with deliberate buffer reuse (~81 MB peak)
  float* ws = (float*)d_ws;
  const size_t IMG = (size_t)BB * CC * HWXY;  // 4,194,304
  float* Y    = ws;                 // [0, 16MB)
  float* T1a  = ws + IMG;           // [16, 32MB)
  float* T1b  = ws + 2 * IMG;       // [32, 48MB)
  float* Z    = ws + 3 * IMG;       // [48, 64MB)
  float* F    = ws + 4 * IMG;       // [64, 80MB)
  float* QKV  = ws + IMG;           // reuses T1a/T1b/Z after scan: 48MB
  float* ATT  = ws + 4 * IMG;       // reuses F after scan: 16MB
  float* qwin = ws + 5 * IMG;       // 65536 floats
  float* kwin = qwin + BB * P2 * CC;
  int*   idx  = (int*)(kwin + BB * P2 * CC);  // 1024 ints

  // mhsa GEMMs (fp32 WMMA)
  gemm_chw<1><<<512, 256, 0, stream>>>(w1a, b1a, x,   T1a);
  gemm_chw<1><<<512, 256, 0, stream>>>(w1b, b1b, x,   T1b);
  gemm_chw<1><<<512, 256, 0, stream>>>(w2a, b2a, T1a, Z);
  gemm_chw<2><<<512, 256, 0, stream>>>(w2b, b2b, T1b, F);

  // direction-split scan along H
  scan_kernel<<<256, 256, 0, stream>>>(Z, F, Y);

  // QKV projection over window tokens
  gemm_qkv<<<1536, 256, 0, stream>>>(qkv_w, qkv_b, Y, QKV);

  // window routing
  winmean<<<256, 256, 0, stream>>>(QKV, qwin, kwin);
  topk_kernel<<<BB * P2, 64, 0, stream>>>(qwin, kwin, idx);

  // top-k gathered window attention
  attn_kernel<<<BB * P2 * HEADS, 256, 0, stream>>>(QKV, idx, ATT);

  // depthwise LePE added to attention output
  lepe_add<<<(NTOK * CC) / 256, 256, 0, stream>>>(QKV, lepe_w, lepe_b, ATT);

  // output projection to (B, C, H, W)
  gemm_out<<<512, 256, 0, stream>>>(wo_w, wo_b, ATT, out);
}